// DetectionLayer_34050500723186
// MI455X (gfx1250) — compile-verified
//
#include <hip/hip_runtime.h>

// YOLO detection-head decode for x:(32,255,76,76) f32 -> (32, 76*76*3, 85) f32.
// Memory-bound: ~377 MB total @ 23.3 TB/s ~= 16 us floor.
//
// Pipeline per (batch,row,half-width) tile:
//   Phase 0: GLOBAL_LOAD_ASYNC_TO_LDS_B64 DMA of the raw channel-major tile
//            into LDS with rows padded to 42 dwords (8B-aligned chunks, and
//            stride 42 mod 64 banks => conflict-free column reads in wave32).
//   Phase 1: read LDS transposed, apply sigmoid/exp/corner math in registers,
//            emit fully coalesced nontemporal b128 stores.

namespace {

constexpr int kNumAttrs = 85;                 // x,y,w,h,conf,80 classes
constexpr int kBpc      = 3;
constexpr int kF        = kBpc * kNumAttrs;   // 255 channels
constexpr int kH        = 76;
constexpr int kW        = 76;
constexpr int kHW       = kH * kW;            // 5776
constexpr int kStride   = 42;                 // padded dwords per LDS row
constexpr float kLog2e   = 1.44269504088896340736f;
constexpr float kInvGrid = 1.0f / 76.0f;

typedef float f32x4 __attribute__((ext_vector_type(4)));

// 0.5 * anchor_wh / image_size, indexed [a*2 + axis]
__device__ __constant__ float kHalfAnchor[6] = {
    10.0f * 0.5f / 608.0f, 13.0f * 0.5f / 608.0f,
    16.0f * 0.5f / 608.0f, 30.0f * 0.5f / 608.0f,
    33.0f * 0.5f / 608.0f, 23.0f * 0.5f / 608.0f};

__device__ __forceinline__ float fast_sigmoid(float v) {
    // 1 / (1 + 2^(-v*log2e)) : v_exp_f32 + v_rcp_f32 (TRANS pipe)
    return __builtin_amdgcn_rcpf(1.0f + __builtin_amdgcn_exp2f(v * -kLog2e));
}
__device__ __forceinline__ float fast_exp(float v) {
    return __builtin_amdgcn_exp2f(v * kLog2e);
}

}  // namespace

__global__ void __launch_bounds__(512)
yolo_decode(const float* __restrict__ in, float* __restrict__ out) {
    __shared__ float A[kF * kStride];   // 255 * 42 * 4 = 42,840 bytes

    const int bid   = blockIdx.x;
    const int halfI = bid & 1;
    const int row   = bid >> 1;
    const int gy    = row % kH;
    const int b     = row / kH;
    const int gx0   = halfI ? 40 : 0;
    const int nx    = halfI ? 36 : 40;

    const float* __restrict__ inBase =
        in + (size_t)b * kF * kHW + (size_t)gy * kW + gx0;
    float* __restrict__ outBase =
        out + ((size_t)b * kHW + (size_t)gy * kW + (size_t)gx0) * kF;

    // ---- Phase 0: async DMA global -> LDS (ASYNCcnt-tracked), b64 chunks.
    // LDS row f lives at byte offset aBase + f*168; chunks stay 8B-aligned.
    const int chunksPerRow = nx >> 1;             // 20 or 18 b64 chunks
    const int nChunks      = kF * chunksPerRow;   // 5100 or 4590
    const uint32_t aBase   = (uint32_t)(uintptr_t)(void*)A;  // LDS byte offset
    for (int k = threadIdx.x; k < nChunks; k += blockDim.x) {
        const int f  = k / chunksPerRow;
        const int xc = k - f * chunksPerRow;
        const uint32_t goff = (uint32_t)f * (uint32_t)(kHW * 4) + (uint32_t)xc * 8u;
        const uint32_t loff = aBase + (uint32_t)f * (uint32_t)(kStride * 4) + (uint32_t)xc * 8u;
        asm volatile("global_load_async_to_lds_b64 %0, %1, %2"
                     :: "v"(loff), "v"(goff), "s"(inBase)
                     : "memory");
    }
    asm volatile("s_wait_asynccnt 0x0" ::: "memory");
    __syncthreads();

    // ---- Phase 1: transposed LDS reads (conflict-free, stride 42 banks),
    // fused transform, coalesced NT b128 stores.
    // Output attrs: [imgx-hw, imgy-hh, imgx+hw, imgy+hh, sig(conf), sig(cls)...]
    const int n2 = (kF * nx) >> 2;
    for (int o4 = threadIdx.x; o4 < n2; o4 += blockDim.x) {
        const int o = o4 << 2;
        f32x4 rv;
#pragma unroll
        for (int j = 0; j < 4; ++j) {
            const int oo  = o + j;
            const int lgx = oo / kF;
            const int f   = oo - lgx * kF;
            const int a   = f / kNumAttrs;
            const int c   = f - a * kNumAttrs;
            const float* __restrict__ col = A + lgx;
            float r;
            if (c >= 4) {
                r = fast_sigmoid(col[f * kStride]);
            } else {
                const int fb   = f - c;        // a*85: start of this box's attrs
                const int axis = c & 1;        // 0 = x/w, 1 = y/h
                float t = fast_sigmoid(col[(fb + axis) * kStride]) * 1.05f - 0.025f;
                float off = axis ? (float)gy : (float)(gx0 + lgx);
                float img = (t + off) * kInvGrid;
                float halfe = fast_exp(col[(fb + 2 + axis) * kStride]) *
                              kHalfAnchor[a * 2 + axis];
                r = (c < 2) ? (img - halfe) : (img + halfe);
            }
            rv[j] = r;
        }
        __builtin_nontemporal_store(rv, (f32x4*)(outBase + o));
    }
}

extern "C" void kernel_launch(void* const* d_in, const int* in_sizes, int n_in,
                              void* d_out, int out_size, void* d_ws, size_t ws_size,
                              hipStream_t stream) {
    (void)in_sizes; (void)n_in; (void)out_size; (void)d_ws; (void)ws_size;
    const float* x = (const float*)d_in[0];
    float* out     = (float*)d_out;

    constexpr int kBatch = 32;
    dim3 grid(kBatch * kH * 2);   // (batch, row, half) tiles = 4864 workgroups
    dim3 block(512);              // 16 wave32s per workgroup
    hipLaunchKernelGGL(yolo_decode, grid, block, 0, stream, x, out);
}